// AttentionOptimizer_30880814858991
// MI455X (gfx1250) — compile-verified
//
#include <hip/hip_runtime.h>

// CDNA5 / gfx1250, wave32.
typedef __attribute__((ext_vector_type(2))) float v2f;
typedef __attribute__((ext_vector_type(4))) float v4f;
typedef __attribute__((ext_vector_type(8))) float v8f;

#define LLAT   20
#define NLAT   8000        // 20^3 lattice sites
#define NPAD   400         // "rest" dimension of each mode-product (20*20)
#define KPAD   32          // padded contraction/output dim (20 -> 32)

// LDS layout (in floats)
#define OFF_K1   0
#define SZ_K1    (KPAD * KPAD)          // 32x32 padded kernel matrix
#define OFF_BUF0 (OFF_K1 + SZ_K1)
#define SZ_BUF   (KPAD * NPAD)          // 32x400 (rows 20..31 zero pad)
#define OFF_BUF1 (OFF_BUF0 + SZ_BUF)
#define OFF_DEN  (OFF_BUF1 + SZ_BUF)
#define OFF_LIN  (OFF_DEN + NLAT)
#define SMEM_FLOATS (OFF_LIN + 32)      // 34656 floats = 138,624 bytes < 320KB/WGP

// One mode product: dst[r*20 + a] = sum_b K1[a,b] * src[b*400 + r]
// (the store permutes the freshly-contracted axis to the fastest position, so
//  the same routine applied 3x contracts x, then y, then z and restores layout).
__device__ __forceinline__ void mode_pass(const float* __restrict__ K1,
                                          const float* __restrict__ src,
                                          float* __restrict__ dst,
                                          int tid) {
    const int wave = tid >> 5;
    const int lane = tid & 31;
    const int ln15 = lane & 15;
    const int kh   = lane >> 4;   // which k-pair this lane supplies (A/B frags)

    // Hoist A fragments: K1 rows are reused by every N-tile of the pass.
    // A (16x4 f32): lanes 0-15 hold M=lane,K={0,1}; lanes 16-31 M=lane-16,K={2,3}
    v2f Alo[8], Ahi[8];
    #pragma unroll
    for (int ks = 0; ks < 8; ++ks) {
        const int kb = ks * 4 + kh * 2;
        Alo[ks].x = K1[ln15 * KPAD + kb];
        Alo[ks].y = K1[ln15 * KPAD + kb + 1];
        Ahi[ks].x = K1[(16 + ln15) * KPAD + kb];
        Ahi[ks].y = K1[(16 + ln15) * KPAD + kb + 1];
    }

    // ---- M-tile 0 (rows 0..15, all valid): 25 N-tiles over 8 waves ----
    for (int nt = wave; nt < 25; nt += 8) {
        const int n = (nt << 4) + ln15;       // global column (r index), < 400
        v8f acc = {0.f, 0.f, 0.f, 0.f, 0.f, 0.f, 0.f, 0.f};
        #pragma unroll
        for (int ks = 0; ks < 8; ++ks) {      // K = 32 in steps of 4
            const int kb = ks * 4 + kh * 2;
            v2f B;                             // B (4x16): lane&15=column, kh=k-pair
            B.x = src[kb * NPAD + n];
            B.y = src[(kb + 1) * NPAD + n];
            acc = __builtin_amdgcn_wmma_f32_16x16x4_f32(
                false, Alo[ks], false, B, (short)0, acc, false, false);
        }
        // D 16x16 f32: lanes 0-15 -> rows v, lanes 16-31 -> rows v+8.
        // 8 contiguous floats per lane; (n*20+m0)*4 is 16B aligned.
        float* p = dst + n * LLAT + (kh ? 8 : 0);
        v4f lo, hi;
        lo.x = acc[0]; lo.y = acc[1]; lo.z = acc[2]; lo.w = acc[3];
        hi.x = acc[4]; hi.y = acc[5]; hi.z = acc[6]; hi.w = acc[7];
        *(v4f*)p       = lo;
        *(v4f*)(p + 4) = hi;
    }

    // ---- M-tile 1 (rows 16..31; only 16..19 valid, rest are exact zeros) ----
    for (int nt = wave; nt < 25; nt += 8) {
        const int n = (nt << 4) + ln15;
        v8f acc = {0.f, 0.f, 0.f, 0.f, 0.f, 0.f, 0.f, 0.f};
        #pragma unroll
        for (int ks = 0; ks < 8; ++ks) {
            const int kb = ks * 4 + kh * 2;
            v2f B;
            B.x = src[kb * NPAD + n];
            B.y = src[(kb + 1) * NPAD + n];
            acc = __builtin_amdgcn_wmma_f32_16x16x4_f32(
                false, Ahi[ks], false, B, (short)0, acc, false, false);
        }
        if (kh == 0) {                         // lanes 0-15 hold rows 16..23
            v4f lo;                            // keep rows 16..19 only
            lo.x = acc[0]; lo.y = acc[1]; lo.z = acc[2]; lo.w = acc[3];
            *(v4f*)(dst + n * LLAT + 16) = lo;
        }                                      // lanes 16-31: rows 24..31, all zero
    }
}

__global__ __launch_bounds__(256)
void lattice_smooth_kernel(const float* __restrict__ grads,
                           const float* __restrict__ spins,
                           const float* __restrict__ pos,
                           const float* __restrict__ noise,
                           float* __restrict__ out) {
    extern __shared__ float smem[];
    const int tid = threadIdx.x;
    const int b   = blockIdx.x;                 // batch
    const float* g = grads + b * NLAT;

    // 1D grid coordinates: pos[:,2] of the first 20 sites == linspace(-1,1,20)
    if (tid < LLAT) smem[OFF_LIN + tid] = pos[3 * tid + 2];
    __syncthreads();

    // Padded 32x32 kernel matrix K1[a,b] = exp(-(lin[a]-lin[b])^2 / 80)
    for (int i = tid; i < SZ_K1; i += 256) {
        const int a = i >> 5, c = i & 31;
        float v = 0.f;
        if (a < LLAT && c < LLAT) {
            const float d = smem[OFF_LIN + a] - smem[OFF_LIN + c];
            v = __expf(-d * d * (1.0f / 80.0f));
        }
        smem[OFF_K1 + i] = v;
    }
    // Zero the K-padding rows (20..31) of both ping-pong buffers (stay zero:
    // every pass writes only indices < 8000).
    for (int i = NLAT + tid; i < SZ_BUF; i += 256) {
        smem[OFF_BUF0 + i] = 0.f;
        smem[OFF_BUF1 + i] = 0.f;
    }
    __syncthreads();

    // chan 0: denominator field a_j = exp(-2|g_j|)
    // chan 1: numerator field   a_j * g_j
    for (int chan = 0; chan < 2; ++chan) {
        for (int n = tid; n < NLAT; n += 256) {
            const float gv = g[n];
            const float aj = __expf(-2.0f * fabsf(gv));
            smem[OFF_BUF0 + n] = chan ? aj * gv : aj;
        }
        __syncthreads();

        mode_pass(smem + OFF_K1, smem + OFF_BUF0, smem + OFF_BUF1, tid);
        __syncthreads();
        mode_pass(smem + OFF_K1, smem + OFF_BUF1, smem + OFF_BUF0, tid);
        __syncthreads();
        mode_pass(smem + OFF_K1, smem + OFF_BUF0, smem + OFF_BUF1, tid);
        __syncthreads();

        if (chan == 0) {
            for (int n = tid; n < NLAT; n += 256)
                smem[OFF_DEN + n] = smem[OFF_BUF1 + n];
            __syncthreads();
        } else {
            // out = spins - LR*(grads + SMOOTH * num/den) + noise
            for (int n = tid; n < NLAT; n += 256) {
                const int idx = b * NLAT + n;
                const float gsm = smem[OFF_BUF1 + n] / smem[OFF_DEN + n];
                out[idx] = spins[idx] - 0.05f * (grads[idx] + 10.0f * gsm) + noise[idx];
            }
        }
    }
}

extern "C" void kernel_launch(void* const* d_in, const int* in_sizes, int n_in,
                              void* d_out, int out_size, void* d_ws, size_t ws_size,
                              hipStream_t stream) {
    const float* grads = (const float*)d_in[0];
    const float* spins = (const float*)d_in[1];
    const float* pos   = (const float*)d_in[2];
    const float* noise = (const float*)d_in[3];
    float* out = (float*)d_out;

    const size_t shmem = SMEM_FLOATS * sizeof(float);
    lattice_smooth_kernel<<<dim3(2), dim3(256), shmem, stream>>>(
        grads, spins, pos, noise, out);
}